// SparseConv_90555090469374
// MI455X (gfx1250) — compile-verified
//
#include <hip/hip_runtime.h>

// ---------------------------------------------------------------------------
// SparseConv chain for MI455X (gfx1250, wave32, WMMA f32 16x16x4).
//
// y[b,o,l] = bias[o] + sum_{c,t} w[o,c,t] * x[b,c,l*st+t]
// K-dim kk = c*KERN + t (matches w memory order (o,c,t) -> contiguous stage,
// and for KERN==2 a (kk,kk+1) B-fragment pair is two adjacent floats -> b64).
// Tail columns (l >= Lout) are computed on clamped addresses and simply not
// stored -- GEMM columns are independent, so no masking in the hot loop.
// Signed 32-bit per-lane element offsets: +1 adjacency is provable (no wrap),
// so each B fragment merges into a single global_load_b64.
// ---------------------------------------------------------------------------

typedef float v2f __attribute__((ext_vector_type(2)));
typedef float v8f __attribute__((ext_vector_type(8)));

#define DCH  256   // channels
#define NMAP 128   // map2d side

// ---- 256 MiB zero fill with b128 stores -----------------------------------
__global__ __launch_bounds__(256) void zero4_kernel(float4* __restrict__ out) {
  size_t i = (size_t)blockIdx.x * 256 + threadIdx.x;
  out[i] = make_float4(0.f, 0.f, 0.f, 0.f);
}

// ---- map2d[b,d,n,n] = x[b,d,n] --------------------------------------------
__global__ __launch_bounds__(256) void diag_kernel(float* __restrict__ out,
                                                   const float* __restrict__ x) {
  unsigned i = blockIdx.x * 256u + threadIdx.x;   // i = (b*256+d)*128 + n
  unsigned n = i & 127u;
  out[(size_t)i * NMAP + n] = x[i];
}

// ---- one conv step --------------------------------------------------------
// grid = (16 m-tiles, 16 batches), block = 128 threads (4 waves).
// Each wave owns 2 N-tiles (32 output positions); block covers l in [0,128).
template <int KERN, int ST>
__global__ __launch_bounds__(128)
void sparseconv_step_kernel(const float* __restrict__ xin,   // (B, D, Lin)
                            const float* __restrict__ w,     // (D, D, KERN)
                            const float* __restrict__ bias,  // (D)
                            float* __restrict__ yout,        // (B, D, Lout)
                            float* __restrict__ map2d,       // (B, D, 128, 128)
                            int Lin, int Lout, int gs, int offset) {
  constexpr int KTOT = KERN * DCH;
  // A-fragment layout wlds[kk][o], pitch 17 dwords -> conflict-free banks.
  __shared__ float wlds[KTOT * 17];

  const int mbase = blockIdx.x << 4;    // 16 output channels per block
  const int b     = blockIdx.y;

  // Cooperative stage, fully coalesced: wlds[kk*17 + o] = w[mbase+o][kk]
  for (int o = 0; o < 16; ++o) {
    const float* wrow = w + (size_t)(mbase + o) * KTOT;
    for (int kk = threadIdx.x; kk < KTOT; kk += 128) {
      wlds[kk * 17 + o] = wrow[kk];
    }
  }
  __syncthreads();

  const int lane  = threadIdx.x & 31;
  const int wave  = threadIdx.x >> 5;
  const int lo    = lane & 15;          // M (A/C) or N (B/C) index
  const int hi    = lane >> 4;          // K-half select for A/B fragments
  const int lbase = wave << 5;          // 2 tiles of 16 output positions

  if (lbase >= Lout) return;            // wave-uniform early out

  // Uniform batch base; per-lane addressing below is signed 32-bit offsets.
  const float* xb = xin + (size_t)b * DCH * Lin;

  // Clamped input positions: tail lanes read column 0 (valid, never stored).
  const int l0 = lbase + lo;
  const int l1 = l0 + 16;
  const int q0 = (l0 < Lout ? l0 : 0) * ST;
  const int q1 = (l1 < Lout ? l1 : 0) * ST;

  v8f acc0 = {};
  v8f acc1 = {};

  if constexpr (KERN == 2) {
    // kk even -> t==0; B pair = x[b,c,q], x[b,c,q+1] (adjacent -> b64 load).
    const int step = 2 * Lin;           // elements per K-step (c += 2)
    int off0 = hi * Lin + q0;           // signed 32-bit per-lane offsets
    int off1 = hi * Lin + q1;
    int ia   = (2 * hi) * 17 + lo;      // A-fragment LDS index
#pragma unroll 8
    for (int k0 = 0; k0 < KTOT; k0 += 4) {
      v2f a;
      a.x = wlds[ia];
      a.y = wlds[ia + 17];
      v2f bb0; bb0.x = xb[off0]; bb0.y = xb[off0 + 1];
      v2f bb1; bb1.x = xb[off1]; bb1.y = xb[off1 + 1];
      acc0 = __builtin_amdgcn_wmma_f32_16x16x4_f32(false, a, false, bb0,
                                                   (short)0, acc0, false, false);
      acc1 = __builtin_amdgcn_wmma_f32_16x16x4_f32(false, a, false, bb1,
                                                   (short)0, acc1, false, false);
      ia   += 4 * 17;
      off0 += step;
      off1 += step;
    }
  } else {
    // KERN == 3: compile-time div/mod by 3 (mul-shift), scalar b32 loads.
#pragma unroll 4
    for (int k0 = 0; k0 < KTOT; k0 += 4) {
      const int kk = k0 + 2 * hi;
      const int c0 = kk / KERN,       t0 = kk - c0 * KERN;
      const int c1 = (kk + 1) / KERN, t1 = (kk + 1) - c1 * KERN;
      v2f a;
      a.x = wlds[kk * 17 + lo];
      a.y = wlds[(kk + 1) * 17 + lo];
      const int e0 = c0 * Lin + t0;     // 32-bit element offsets
      const int e1 = c1 * Lin + t1;
      v2f bb0; bb0.x = xb[e0 + q0]; bb0.y = xb[e1 + q0];
      v2f bb1; bb1.x = xb[e0 + q1]; bb1.y = xb[e1 + q1];
      acc0 = __builtin_amdgcn_wmma_f32_16x16x4_f32(false, a, false, bb0,
                                                   (short)0, acc0, false, false);
      acc1 = __builtin_amdgcn_wmma_f32_16x16x4_f32(false, a, false, bb1,
                                                   (short)0, acc1, false, false);
    }
  }

  // Bias (C/D layout: VGPR r holds rows r (lanes 0-15) and r+8 (lanes 16-31))
  float bvals[8];
#pragma unroll
  for (int r = 0; r < 8; ++r) bvals[r] = bias[mbase + r + 8 * hi];

  // 32-bit store indexing off uniform bases (yout, map2d).
  const int obase = mbase + 8 * hi;
  const int ybB   = b * (DCH * Lout);        // < 2^20
  const int mbB   = b * (DCH * NMAP * NMAP); // < 2^27

  // Store tile 0
  if (l0 < Lout) {
    const int ii = l0 * gs, jj = offset + l0 * gs;
    int yi = ybB + obase * Lout + l0;
    int mi = mbB + obase * (NMAP * NMAP) + ii * NMAP + jj;
#pragma unroll
    for (int r = 0; r < 8; ++r) {
      const float v = acc0[r] + bvals[r];
      yout[yi]  = v;
      map2d[mi] = v;
      yi += Lout;
      mi += NMAP * NMAP;
    }
  }
  // Store tile 1
  if (l1 < Lout) {
    const int ii = l1 * gs, jj = offset + l1 * gs;
    int yi = ybB + obase * Lout + l1;
    int mi = mbB + obase * (NMAP * NMAP) + ii * NMAP + jj;
#pragma unroll
    for (int r = 0; r < 8; ++r) {
      const float v = acc1[r] + bvals[r];
      yout[yi]  = v;
      map2d[mi] = v;
      yi += Lout;
      mi += NMAP * NMAP;
    }
  }
}

// ---------------------------------------------------------------------------
extern "C" void kernel_launch(void* const* d_in, const int* in_sizes, int n_in,
                              void* d_out, int out_size, void* d_ws, size_t ws_size,
                              hipStream_t stream) {
  const float* x  = (const float*)d_in[0];   // (16, 256, 128)
  const float* w2 = (const float*)d_in[1];   // (29, 256, 256, 2)
  const float* b2 = (const float*)d_in[2];   // (29, 256)
  const float* w3 = (const float*)d_in[3];   // (2, 256, 256, 3)
  const float* b3 = (const float*)d_in[4];   // (2, 256)
  float* out = (float*)d_out;                // (16, 256, 128, 128)

  // Ping-pong activation buffers in workspace (2 MiB each).
  const size_t bufElems = (size_t)16 * DCH * NMAP;
  float* buf0 = (float*)d_ws;
  float* buf1 = buf0 + bufElems;

  // 1) zero the 256 MiB map2d: 67,108,864 floats = 16,777,216 float4
  zero4_kernel<<<65536, 256, 0, stream>>>((float4*)out);
  // 2) diagonal = x
  diag_kernel<<<2048, 256, 0, stream>>>(out, x);

  // 3) the 31-step conv chain (stream-ordered dependencies)
  const int counts[3] = {15, 8, 8};
  int L = NMAP, gs = 1, off = 0, c2 = 0, c3 = 0;
  const float* cur = x;
  float* nxt = buf0;

  for (int ci = 0; ci < 3; ++ci) {
    for (int k = 0; k < counts[ci]; ++k) {
      off += gs;
      const bool down = (ci > 0 && k == 0);
      dim3 grid(16, 16);  // 16 m-tiles x 16 batches

      if (!down) {
        const int Lout = L - 1;
        const float* wp = w2 + (size_t)c2 * DCH * DCH * 2;
        const float* bp = b2 + (size_t)c2 * DCH;
        ++c2;
        sparseconv_step_kernel<2, 1><<<grid, 128, 0, stream>>>(
            cur, wp, bp, nxt, out, L, Lout, gs, off);
        L = Lout;
      } else {
        const int Lout = (L - 3) / 2 + 1;
        const float* wp = w3 + (size_t)c3 * DCH * DCH * 3;
        const float* bp = b3 + (size_t)c3 * DCH;
        ++c3;
        sparseconv_step_kernel<3, 2><<<grid, 128, 0, stream>>>(
            cur, wp, bp, nxt, out, L, Lout, gs, off);
        L = Lout;
      }

      cur = nxt;
      nxt = (nxt == buf0) ? buf1 : buf0;
    }
    gs *= 2;
  }
}